// Attention_3710851743764
// MI455X (gfx1250) — compile-verified
//
#include <hip/hip_runtime.h>
#include <math.h>

typedef __attribute__((ext_vector_type(2))) float v2f;
typedef __attribute__((ext_vector_type(8))) float v8f;

#define DIM 2048
#define NH 16
#define NKV 4
#define HD 128
#define BATCH 2
#define SEQ 2048
#define QKV_N 3072          // (NH + 2*NKV) * HD
#define TOKENS (BATCH*SEQ)  // 4096
#define RMS_EPS 1.1920928955078125e-7f

// ---- CDNA5 async copy engine: global -> LDS direct, tracked by ASYNCcnt ----
__device__ __forceinline__ void async_copy_b128(void* lds_dst, const void* gsrc) {
  unsigned loff = (unsigned)(unsigned long long)lds_dst;  // low 32 bits = LDS byte offset
  asm volatile("global_load_async_to_lds_b128 %0, %1, off"
               :: "v"(loff), "v"(gsrc)
               : "memory");
}
__device__ __forceinline__ void wait_async0() {
  asm volatile("s_wait_asynccnt 0x0" ::: "memory");
}

// =====================================================================
// GEMM: C[M,N] = A[M,K] @ W[N,K]^T   (fp32, V_WMMA_F32_16X16X4_F32)
// Block tile 64(M) x 128(N), K-chunk 32, 8 waves each doing 32x32.
// Double-buffered LDS filled by the async copy engine.
// =====================================================================
#define BM 64
#define BN 128
#define BK 32
#define LDSS 36   // padded row stride (floats): conflict-free ds_load_b64

__global__ __launch_bounds__(256) void gemm_f32_wmma(
    const float* __restrict__ A, const float* __restrict__ W,
    float* __restrict__ C, int M, int N, int K)
{
  __shared__ float As[2][BM * LDSS];
  __shared__ float Bs[2][BN * LDSS];

  const int tid  = threadIdx.x;
  const int lane = tid & 31;
  const int l15  = lane & 15;
  const int hi   = lane >> 4;           // 0: k0..k0+1, 1: k0+2..k0+3
  const int wid  = tid >> 5;            // 8 waves
  const int wm   = wid & 1;             // 2 waves along M
  const int wn   = wid >> 1;            // 4 waves along N
  const int m0   = blockIdx.y * BM;
  const int n0   = blockIdx.x * BN;

  auto stage = [&](int buf, int kc) {
#pragma unroll
    for (int p = 0; p < 2; ++p) {                 // A: 64 rows x 8 float4
      int idx = tid + p * 256;
      int row = idx >> 3, c4 = (idx & 7) << 2;
      async_copy_b128(&As[buf][row * LDSS + c4],
                      &A[(size_t)(m0 + row) * K + kc + c4]);
    }
#pragma unroll
    for (int p = 0; p < 4; ++p) {                 // W: 128 rows x 8 float4
      int idx = tid + p * 256;
      int row = idx >> 3, c4 = (idx & 7) << 2;
      async_copy_b128(&Bs[buf][row * LDSS + c4],
                      &W[(size_t)(n0 + row) * K + kc + c4]);
    }
  };

  v8f acc[2][2] = {};

  stage(0, 0);
  wait_async0();
  __syncthreads();

  int buf = 0;
  for (int kc = 0; kc < K; kc += BK) {
    if (kc + BK < K) stage(buf ^ 1, kc + BK);     // overlap fill with compute

#pragma unroll
    for (int k0 = 0; k0 < BK; k0 += 4) {
      v2f a0 = *(const v2f*)&As[buf][(wm * 32 + l15) * LDSS + k0 + 2 * hi];
      v2f a1 = *(const v2f*)&As[buf][(wm * 32 + 16 + l15) * LDSS + k0 + 2 * hi];
      v2f b0 = *(const v2f*)&Bs[buf][(wn * 32 + l15) * LDSS + k0 + 2 * hi];
      v2f b1 = *(const v2f*)&Bs[buf][(wn * 32 + 16 + l15) * LDSS + k0 + 2 * hi];
      acc[0][0] = __builtin_amdgcn_wmma_f32_16x16x4_f32(false, a0, false, b0, (short)0, acc[0][0], false, false);
      acc[0][1] = __builtin_amdgcn_wmma_f32_16x16x4_f32(false, a0, false, b1, (short)0, acc[0][1], false, false);
      acc[1][0] = __builtin_amdgcn_wmma_f32_16x16x4_f32(false, a1, false, b0, (short)0, acc[1][0], false, false);
      acc[1][1] = __builtin_amdgcn_wmma_f32_16x16x4_f32(false, a1, false, b1, (short)0, acc[1][1], false, false);
    }

    wait_async0();        // my async fills of buf^1 have landed in LDS
    __syncthreads();      // everyone done computing on buf + filling buf^1
    buf ^= 1;
  }

  // ---- epilogue: C/D layout: VGPR r -> (row r, lanes0-15) / (row r+8, lanes16-31)
#pragma unroll
  for (int i = 0; i < 2; ++i)
#pragma unroll
    for (int j = 0; j < 2; ++j)
#pragma unroll
      for (int r = 0; r < 8; ++r) {
        int row = m0 + wm * 32 + i * 16 + r + 8 * hi;
        int col = n0 + wn * 32 + j * 16 + l15;
        C[(size_t)row * N + col] = acc[i][j][r];
      }
}

// =====================================================================
// RMSNorm (q & k heads) + interleaved-pair RoPE, in place in qkv buffer.
// One wave per (token, head). 128 dims = 4 floats/lane.
// =====================================================================
__global__ __launch_bounds__(256) void rmsnorm_rope(
    float* __restrict__ qkv,
    const float* __restrict__ qw, const float* __restrict__ kw,
    const float* __restrict__ fcos, const float* __restrict__ fsin)
{
  const int tid  = threadIdx.x;
  const int lane = tid & 31;
  const int wid  = tid >> 5;
  const int gw   = blockIdx.x * 8 + wid;
  const int token = gw / (NH + NKV);
  const int head  = gw % (NH + NKV);
  const int s     = token & (SEQ - 1);

  size_t base;
  const float* w;
  if (head < NH) { base = (size_t)token * QKV_N + head * HD;                  w = qw; }
  else           { base = (size_t)token * QKV_N + NH * HD + (head - NH) * HD; w = kw; }

  float4 x = *(const float4*)&qkv[base + lane * 4];
  float ss = x.x * x.x + x.y * x.y + x.z * x.z + x.w * x.w;
  ss += __shfl_xor(ss, 1, 32);
  ss += __shfl_xor(ss, 2, 32);
  ss += __shfl_xor(ss, 4, 32);
  ss += __shfl_xor(ss, 8, 32);
  ss += __shfl_xor(ss, 16, 32);
  const float rinv = rsqrtf(ss * (1.0f / HD) + RMS_EPS);

  float4 wv = *(const float4*)&w[lane * 4];
  float4 c  = *(const float4*)&fcos[(size_t)s * HD + lane * 4];
  float4 sn = *(const float4*)&fsin[(size_t)s * HD + lane * 4];

  float n0 = x.x * rinv * wv.x, n1 = x.y * rinv * wv.y;
  float n2 = x.z * rinv * wv.z, n3 = x.w * rinv * wv.w;
  float4 o;
  o.x = n0 * c.x - n1 * sn.x;
  o.y = n1 * c.y + n0 * sn.y;
  o.z = n2 * c.z - n3 * sn.z;
  o.w = n3 * c.w + n2 * sn.w;
  *(float4*)&qkv[base + lane * 4] = o;
}

// =====================================================================
// Flash attention, fp32 WMMA. One wave owns a 16-row Q tile; 8 waves per
// block share double-buffered K/V 16x128 LDS tiles staged by the async
// copy engine; online softmax; P re-layout via per-wave LDS tile.
// =====================================================================
#define KVS 132   // K/V LDS row stride (floats) -> 16B-aligned float4 rows
#define PST 18    // P tile row stride (even -> ds_load_b64 aligned, conflict-free)

__global__ __launch_bounds__(256) void attn_flash(
    const float* __restrict__ qkv, float* __restrict__ out)
{
  __shared__ float Ks[2][16 * KVS];
  __shared__ float Vs[2][16 * KVS];
  __shared__ float Ps[8 * 16 * PST];

  const int tid  = threadIdx.x;
  const int lane = tid & 31;
  const int l15  = lane & 15;
  const int hi   = lane >> 4;
  const int wid  = tid >> 5;
  const int b  = blockIdx.z;
  const int h  = blockIdx.y;
  const int q0 = blockIdx.x * 128;
  const int kv = h >> 2;                 // GQA: 4 q heads per kv head
  float* Pm = &Ps[wid * 16 * PST];

  auto stage = [&](int buf, int j0) {
#pragma unroll
    for (int p = 0; p < 2; ++p) {        // 512 float4 per tile, 2/thread
      int idx = tid + p * 256;
      int row = idx >> 5, c4 = (idx & 31) << 2;
      size_t g = (size_t)(b * SEQ + j0 + row) * QKV_N + kv * HD;
      async_copy_b128(&Ks[buf][row * KVS + c4], &qkv[g + NH * HD + c4]);
      async_copy_b128(&Vs[buf][row * KVS + c4], &qkv[g + NH * HD + NKV * HD + c4]);
    }
  };

  // ---- load Q tile as 32 A-fragments (16 rows x 128 dims) ----
  const size_t qbase = (size_t)(b * SEQ + q0 + wid * 16 + l15) * QKV_N + h * HD;
  v2f qf[32];
#pragma unroll
  for (int kk = 0; kk < 32; ++kk)
    qf[kk] = *(const v2f*)&qkv[qbase + kk * 4 + 2 * hi];

  v8f o[8] = {};
  float mrow[8], lrow[8];
#pragma unroll
  for (int r = 0; r < 8; ++r) { mrow[r] = -__builtin_inff(); lrow[r] = 0.0f; }

  const float scale = 0.08838834764831845f;   // 1/sqrt(128)

  stage(0, 0);
  wait_async0();
  __syncthreads();

  int buf = 0;
  for (int j0 = 0; j0 < SEQ; j0 += 16) {
    if (j0 + 16 < SEQ) stage(buf ^ 1, j0 + 16);   // prefetch next K/V block

    // ---- scores: S(16x16) = Q(16x128) @ K^T, 32 chained WMMAs ----
    v8f sc = {};
#pragma unroll
    for (int kk = 0; kk < 32; ++kk) {
      v2f bf = *(const v2f*)&Ks[buf][l15 * KVS + kk * 4 + 2 * hi];
      sc = __builtin_amdgcn_wmma_f32_16x16x4_f32(false, qf[kk], false, bf, (short)0, sc, false, false);
    }

    // ---- online softmax (row = r + 8*hi lives in 16-lane half-wave) ----
    float p[8], alpha[8];
#pragma unroll
    for (int r = 0; r < 8; ++r) {
      float x = sc[r] * scale;
      float xm = x;
      xm = fmaxf(xm, __shfl_xor(xm, 1, 32));
      xm = fmaxf(xm, __shfl_xor(xm, 2, 32));
      xm = fmaxf(xm, __shfl_xor(xm, 4, 32));
      xm = fmaxf(xm, __shfl_xor(xm, 8, 32));
      float mn = fmaxf(mrow[r], xm);
      alpha[r] = __expf(mrow[r] - mn);
      p[r] = __expf(x - mn);
      float rs = p[r];
      rs += __shfl_xor(rs, 1, 32);
      rs += __shfl_xor(rs, 2, 32);
      rs += __shfl_xor(rs, 4, 32);
      rs += __shfl_xor(rs, 8, 32);
      lrow[r] = lrow[r] * alpha[r] + rs;
      mrow[r] = mn;
    }
#pragma unroll
    for (int t = 0; t < 8; ++t)
#pragma unroll
      for (int r = 0; r < 8; ++r)
        o[t][r] *= alpha[r];

    // ---- re-layout P (C-layout -> A-layout) via per-wave LDS tile ----
#pragma unroll
    for (int r = 0; r < 8; ++r)
      Pm[(r + 8 * hi) * PST + l15] = p[r];
    // same-wave DS ops are in-order: RAW through LDS is safe

    // ---- O += P(16x16) @ V(16x128): 4 K-steps x 8 output tiles ----
#pragma unroll
    for (int k4 = 0; k4 < 4; ++k4) {
      v2f af = *(const v2f*)&Pm[l15 * PST + k4 * 4 + 2 * hi];
#pragma unroll
      for (int t = 0; t < 8; ++t) {
        v2f bf;
        bf.x = Vs[buf][(k4 * 4 + 2 * hi) * KVS + t * 16 + l15];
        bf.y = Vs[buf][(k4 * 4 + 2 * hi + 1) * KVS + t * 16 + l15];
        o[t] = __builtin_amdgcn_wmma_f32_16x16x4_f32(false, af, false, bf, (short)0, o[t], false, false);
      }
    }

    wait_async0();        // my async fills of buf^1 landed
    __syncthreads();      // all waves done with buf + all fills visible
    buf ^= 1;
  }

  // ---- normalize & write [b, s, h*HD] so the final GEMM reads [4096,2048]
#pragma unroll
  for (int r = 0; r < 8; ++r) {
    float inv = 1.0f / lrow[r];
    size_t row = (size_t)(b * SEQ + q0 + wid * 16 + r + 8 * hi);
#pragma unroll
    for (int t = 0; t < 8; ++t)
      out[row * DIM + h * HD + t * 16 + l15] = o[t][r] * inv;
  }
}

// =====================================================================
extern "C" void kernel_launch(void* const* d_in, const int* in_sizes, int n_in,
                              void* d_out, int out_size, void* d_ws, size_t ws_size,
                              hipStream_t stream)
{
  const float* x    = (const float*)d_in[0];
  const float* wqkv = (const float*)d_in[1];
  const float* wo   = (const float*)d_in[2];
  const float* qw   = (const float*)d_in[3];
  const float* kw   = (const float*)d_in[4];
  const float* fc   = (const float*)d_in[5];
  const float* fs   = (const float*)d_in[6];
  float* out  = (float*)d_out;

  float* qkv  = (float*)d_ws;                       // 4096*3072 f32 (48 MB)
  float* attn = qkv + (size_t)TOKENS * QKV_N;       // 4096*2048 f32 (32 MB)

  // 1) QKV projection
  gemm_f32_wmma<<<dim3(QKV_N / BN, TOKENS / BM), 256, 0, stream>>>(
      x, wqkv, qkv, TOKENS, QKV_N, DIM);
  // 2) RMSNorm + RoPE on q/k heads (in place)
  rmsnorm_rope<<<TOKENS * (NH + NKV) / 8, 256, 0, stream>>>(qkv, qw, kw, fc, fs);
  // 3) Flash attention
  attn_flash<<<dim3(SEQ / 128, NH, BATCH), 256, 0, stream>>>(qkv, attn);
  // 4) Output projection
  gemm_f32_wmma<<<dim3(DIM / BN, TOKENS / BM), 256, 0, stream>>>(
      attn, wo, out, TOKENS, DIM, DIM);
}